// MixtureOfExpertsNet_4870492914002
// MI455X (gfx1250) — compile-verified
//
#include <hip/hip_runtime.h>
#include <math.h>

// MoE per-row kernel for MI455X (gfx1250, wave32).
// Roofline: 20 B/row @ 23.3 TB/s -> ~7.2us memory; ~190 fp32 VALU ops/row
// dominates -> minimize op count:
//   relu-sum rewrite:  sum_h W2*relu(W1*x+b1) = A*x + B + sum_h c_h*|x - t_h|
//                      (v_sub + v_fma with free |abs| source modifier)
//   softmax cancel:    pred = sum(m*e^l*adj) / sum(m*e^l)  (normalizer cancels)
//   exp2 domain:       gating weights pre-scaled by log2(e) -> bare v_exp_f32
// Weight scalars are pinned into VGPRs at load time (inline-asm "+v"), making
// the whole constant transform VALU/VGPR-resident (no SGPR spills; preamble
// moves dual-issue as v_dual_mov). Row stream uses NT-hint b128 loads / b32
// stores, a 1-deep software pipeline, plus global_prefetch ~4 iters ahead.

typedef float v4f __attribute__((ext_vector_type(4)));

#define NE 4
#define NH 16

#if __has_builtin(__builtin_amdgcn_exp2f)
#define EXP2F(v) __builtin_amdgcn_exp2f(v)
#else
#define EXP2F(v) __expf((v) * 0.6931471805599453f)
#endif

#define LOG2E 1.4426950408889634f

// Load a (wave-uniform) scalar and force the value into the VGPR register
// class; inline-asm results are divergent to the compiler, so all downstream
// arithmetic stays on the VALU in VGPRs instead of piling up SGPR pressure.
__device__ __forceinline__ float vload(const float* __restrict__ p) {
    float v = *p;
    asm("" : "+v"(v));
    return v;
}

__global__ __launch_bounds__(256) void moe_fused_kernel(
    const float* __restrict__ x,
    const float* __restrict__ Wg,
    const float* __restrict__ bg,
    const float* __restrict__ W1,
    const float* __restrict__ b1,
    const float* __restrict__ W2,
    const float* __restrict__ b2,
    float* __restrict__ out,
    int B)
{
    const int tid    = blockIdx.x * blockDim.x + threadIdx.x;
    const int stride = gridDim.x * blockDim.x;

    // ---- one-time constant transform (branchless, VALU/VGPR-resident;
    // amortized over ~32 grid-stride rows) ----
    float tA[NE], tB[NE];        // linear part: A*x + B
    float tc[NE][NH];            // c_h = 0.5*W2*|W1|
    float tt[NE][NH];            // t_h = -b1/W1 (breakpoint)
    #pragma unroll
    for (int e = 0; e < NE; ++e) {
        float a = 0.0f, b = 0.0f;
        #pragma unroll
        for (int h = 0; h < NH; ++h) {
            float w1  = vload(&W1[e * NH + h]);
            float bb  = vload(&b1[e * NH + h]);
            float hw2 = 0.5f * vload(&W2[e * NH + h]);
            a = fmaf(hw2, w1, a);
            b = fmaf(hw2, bb, b);
            float r  = __builtin_amdgcn_rcpf(w1);     // v_rcp_f32, no div expansion
            bool  nz = (w1 != 0.0f);
            tc[e][h] = hw2 * fabsf(w1);               // == 0 when w1 == 0
            tt[e][h] = nz ? (-bb * r) : 0.0f;         // select, not *mask (inf*0=NaN)
            b = fmaf(hw2, nz ? 0.0f : fabsf(bb), b);  // fold W2*relu(b1) if w1==0
        }
        tA[e] = a;
        tB[e] = b + vload(&b2[e]);
    }
    // gating constants pre-scaled into the exp2 domain
    float wgc[NE][NE], bgc[NE];
    #pragma unroll
    for (int e = 0; e < NE; ++e) {
        bgc[e] = vload(&bg[e]) * LOG2E;
        #pragma unroll
        for (int k = 0; k < NE; ++k) wgc[e][k] = vload(&Wg[e * NE + k]) * LOG2E;
    }

    const v4f* __restrict__ x4 = (const v4f*)x;

    int i = tid;
    if (i >= B) return;
    v4f xv = __builtin_nontemporal_load(&x4[i]);   // global_load_b128 th:NT

    #pragma unroll 1
    for (; i < B; i += stride) {
        // deep prefetch of the row stream (speculative global_prefetch_b8:
        // no LOADcnt, safely dropped past end-of-buffer per ISA)
        __builtin_prefetch(&x4[i + 4 * stride], 0, 1);

        // 1-deep software pipeline: next row's load before this row's compute
        int inext = i + stride;
        v4f xn = xv;
        if (inext < B) xn = __builtin_nontemporal_load(&x4[inext]);

        float xr[NE] = {xv.x, xv.y, xv.z, xv.w};
        float xf[NE];
        bool  mk[NE];
        #pragma unroll
        for (int e = 0; e < NE; ++e) {
            mk[e] = (xr[e] == xr[e]);          // !isnan
            xf[e] = mk[e] ? xr[e] : 0.0f;
        }

        // per-expert MLP: adjusted_e = relu(A*x + B + sum_h c_h*|x - t_h|)
        float adj[NE];
        #pragma unroll
        for (int e = 0; e < NE; ++e) {
            float xe  = xf[e];
            float acc = fmaf(tA[e], xe, tB[e]);
            #pragma unroll
            for (int h = 0; h < NH; ++h) {
                acc = fmaf(tc[e][h], fabsf(xe - tt[e][h]), acc);
            }
            adj[e] = fmaxf(acc, 0.0f);
        }

        // gating logits in exp2 domain; softmax normalizer cancels.
        float l[NE];
        #pragma unroll
        for (int e = 0; e < NE; ++e) {
            float acc = bgc[e];
            #pragma unroll
            for (int k = 0; k < NE; ++k) acc = fmaf(wgc[e][k], xf[k], acc);
            l[e] = acc;
        }
        float mx = fmaxf(fmaxf(l[0], l[1]), fmaxf(l[2], l[3]));

        float den = 0.0f, num = 0.0f;
        #pragma unroll
        for (int e = 0; e < NE; ++e) {
            float ex = EXP2F(l[e] - mx);       // bare v_exp_f32 (trans)
            ex = mk[e] ? ex : 0.0f;
            den += ex;
            num = fmaf(ex, adj[e], num);
        }

        float q   = num * __builtin_amdgcn_rcpf(den);
        float res = (den > 0.0f) ? q : __builtin_nanf("");
        __builtin_nontemporal_store(res, &out[i]);  // global_store_b32 th:NT

        xv = xn;
    }
}

extern "C" void kernel_launch(void* const* d_in, const int* in_sizes, int n_in,
                              void* d_out, int out_size, void* d_ws, size_t ws_size,
                              hipStream_t stream) {
    const float* x  = (const float*)d_in[0];
    const float* Wg = (const float*)d_in[1];
    const float* bg = (const float*)d_in[2];
    const float* W1 = (const float*)d_in[3];
    const float* b1 = (const float*)d_in[4];
    const float* W2 = (const float*)d_in[5];
    const float* b2 = (const float*)d_in[6];
    float* out = (float*)d_out;

    const int B = in_sizes[0] / NE;     // rows
    const int threads = 256;            // 8 wave32 waves per block
    const int blocks  = 1024;           // grid-stride: ~32 rows/thread at B=8.4M

    hipLaunchKernelGGL(moe_fused_kernel, dim3(blocks), dim3(threads), 0, stream,
                       x, Wg, bg, W1, b1, W2, b2, out, B);
}